// LeafOnlyNet_68307159876026
// MI455X (gfx1250) — compile-verified
//
#include <hip/hip_runtime.h>

// Fused block-sparse graph attention for MI455X (gfx1250, wave32).
// One workgroup (128 threads = 4 wave32) per 32-token block.
// All dense GEMMs use V_WMMA_F32_16X16X4_F32 (fp32-exact).

typedef __attribute__((ext_vector_type(2))) float v2f;
typedef __attribute__((ext_vector_type(8))) float v8f;

#define BSZ   32      // block size (queries per block)
#define CDIM  128     // channels
#define NHEAD 4
#define HDIM  32
#define NBLK  4096
#define LSTR  129     // LDS row stride (pad to dodge 64-bank conflicts)
#define NEGV  (-1e30f)

__device__ __forceinline__ v8f wmma4(v2f a, v2f b, v8f c) {
    // D = A(16x4,f32) * B(4x16,f32) + C(16x16,f32)
    return __builtin_amdgcn_wmma_f32_16x16x4_f32(false, a, false, b,
                                                 (short)0, c, false, false);
}

// Compute one 16x16 output tile: A rows [rowBase..rowBase+15] of LDS (stride LSTR),
// B = weight row `wrow` (weight is [out,in] row-major, out column = fixed per lane),
// accumulate over K = CDIM in steps of 4.
__device__ __forceinline__ v8f tile_gemm(const float* __restrict__ sArow, int rowBase,
                                         const float* __restrict__ wrow,
                                         int lane, float bias) {
    const int mLane = lane & 15;            // M for A, N for B/C/D
    const int kOff  = (lane >> 4) << 1;     // lanes 0-15 -> K {0,1}; 16-31 -> K {2,3}
    v8f acc;
    #pragma unroll
    for (int r = 0; r < 8; ++r) acc[r] = bias;
    #pragma unroll
    for (int k0 = 0; k0 < CDIM; k0 += 4) {
        v2f a, b;
        a.x = sArow[(rowBase + mLane) * LSTR + k0 + kOff];
        a.y = sArow[(rowBase + mLane) * LSTR + k0 + kOff + 1];
        b.x = wrow[k0 + kOff];
        b.y = wrow[k0 + kOff + 1];
        acc = wmma4(a, b, acc);
    }
    return acc;
}

__device__ __forceinline__ void tile_store_lds(float* __restrict__ dst, v8f acc,
                                               int rowBase, int colBase, int lane) {
    const int mLane = lane & 15;
    const int rOff  = (lane < 16) ? 0 : 8;  // lanes 0-15: rows 0-7; 16-31: rows 8-15
    #pragma unroll
    for (int r = 0; r < 8; ++r)
        dst[(rowBase + rOff + r) * LSTR + colBase + mLane] = acc[r];
}

__launch_bounds__(128)
__global__ void leafattn_kernel(const float* __restrict__ x,
                                const float* __restrict__ edge_feats,
                                const float* __restrict__ Wqkv,
                                const float* __restrict__ bqkv,
                                const float* __restrict__ Wproj,
                                const float* __restrict__ bproj,
                                const float* __restrict__ Wgate,
                                const float* __restrict__ bgate,
                                const int*   __restrict__ attn_mask,
                                float*       __restrict__ out)
{
    __shared__ float sX[BSZ * LSTR];        // token features
    __shared__ float sQ[BSZ * LSTR];        // Q, later attn output
    __shared__ float sKV[(BSZ + 1) * LSTR]; // K (row 32 = k_global), later V

    const int blk  = blockIdx.x;
    const int tid  = threadIdx.x;
    const int lane = tid & 31;
    const int wave = tid >> 5;

    // ---------------- Phase 0: load X block into LDS ----------------
    {
        const float* xblk = x + (size_t)blk * BSZ * CDIM;
        for (int i = tid; i < BSZ * CDIM; i += 128) {
            int r = i >> 7, c = i & 127;
            sX[r * LSTR + c] = xblk[i];
        }
    }
    __syncthreads();

    // ---------------- Phase 1: Q = X Wq^T + bq ; K = X Wk^T + bk (WMMA) ----
    // 32 tiles (16 Q + 16 K) over [32 rows x 128 cols] split across 4 waves.
    for (int job = wave; job < 32; job += 4) {
        const int isK     = job >> 4;
        const int tileIdx = job & 15;
        const int rowBase = (tileIdx >> 3) << 4;
        const int colBase = (tileIdx & 7) << 4;
        const int outBase = isK ? CDIM : 0;          // Wqkv row offset (out,in layout)
        const int nCol    = colBase + (lane & 15);
        const float bias  = bqkv[outBase + nCol];
        const float* wrow = Wqkv + (size_t)(outBase + nCol) * CDIM;
        v8f acc = tile_gemm(sX, rowBase, wrow, lane, bias);
        tile_store_lds(isK ? sKV : sQ, acc, rowBase, colBase, lane);
    }
    __syncthreads();

    // k_global = column mean of K rows (== K(mean_x) since projection is affine)
    {
        float s = 0.f;
        #pragma unroll
        for (int r = 0; r < BSZ; ++r) s += sKV[r * LSTR + tid];
        sKV[BSZ * LSTR + tid] = s * (1.0f / BSZ);
    }
    __syncthreads();

    // ---------------- Phase 2: scores, bias, mask, softmax, gate ----------
    const int m = tid >> 2;                 // query row 0..31
    const int h = tid & 3;                  // head 0..3
    const float scale = 0.17677669529663687f;   // 1/sqrt(HDIM)
    const float wg0 = Wgate[h * 4 + 0], wg1 = Wgate[h * 4 + 1];
    const float wg2 = Wgate[h * 4 + 2], wg3 = Wgate[h * 4 + 3];
    const float bg  = bgate[h];

    float comb[BSZ + 1];                    // scores -> probs -> probs+gate
    float lew [BSZ + 1];
    {
        const float* qrow = &sQ[m * LSTR + h * HDIM];
        float qreg[HDIM];
        #pragma unroll
        for (int d = 0; d < HDIM; ++d) qreg[d] = qrow[d];

        const int mbase = (blk * BSZ + m) * (BSZ + 1);
        float mx = -3.4e38f;
        #pragma unroll
        for (int j = 0; j <= BSZ; ++j) {
            const float* krow = &sKV[j * LSTR + h * HDIM];
            float dot = 0.f;
            #pragma unroll
            for (int d = 0; d < HDIM; ++d) dot += qreg[d] * krow[d];
            float e0, e1, e2, e3;
            if (j == BSZ || j == m) { e0 = e1 = e2 = 0.f; e3 = 1.f; }
            else {
                const float* ef = edge_feats + (size_t)(mbase + j) * 4;
                e0 = ef[0]; e1 = ef[1]; e2 = ef[2]; e3 = ef[3];
            }
            float sc = dot * scale + e3;
            const int mv = attn_mask[mbase + j];
            if (!mv) sc = NEGV;
            comb[j] = sc;
            lew[j]  = mv ? (e0 * wg0 + e1 * wg1 + e2 * wg2 + e3 * wg3 + bg) : 0.f;
            mx = fmaxf(mx, sc);
        }
        float sum = 0.f;
        #pragma unroll
        for (int j = 0; j <= BSZ; ++j) { comb[j] = __expf(comb[j] - mx); sum += comb[j]; }
        const float inv = 1.0f / sum;
        #pragma unroll
        for (int j = 0; j <= BSZ; ++j) comb[j] = comb[j] * inv + lew[j];
    }
    __syncthreads();   // Q and K no longer needed

    // ---------------- Phase 3: V = X Wv^T + bv (WMMA, overwrites K) -------
    for (int t = wave; t < 16; t += 4) {
        const int rowBase = (t >> 3) << 4;
        const int colBase = (t & 7) << 4;
        const int nCol    = colBase + (lane & 15);
        const float bias  = bqkv[2 * CDIM + nCol];
        const float* wrow = Wqkv + (size_t)(2 * CDIM + nCol) * CDIM;
        v8f acc = tile_gemm(sX, rowBase, wrow, lane, bias);
        tile_store_lds(sKV, acc, rowBase, colBase, lane);
    }
    __syncthreads();

    // v_global = column mean of V rows
    {
        float s = 0.f;
        #pragma unroll
        for (int r = 0; r < BSZ; ++r) s += sKV[r * LSTR + tid];
        sKV[BSZ * LSTR + tid] = s * (1.0f / BSZ);
    }
    __syncthreads();

    // ---------------- Phase 4: attn_out = combined . V (into sQ) ----------
    {
        #pragma unroll
        for (int d = 0; d < HDIM; ++d) {
            float acc = 0.f;
            #pragma unroll
            for (int j = 0; j <= BSZ; ++j) acc += comb[j] * sKV[j * LSTR + h * HDIM + d];
            sQ[m * LSTR + h * HDIM + d] = acc;
        }
    }
    __syncthreads();

    // ---------------- Phase 5: out = attn_out Wproj^T + bproj (WMMA) ------
    for (int t = wave; t < 16; t += 4) {
        const int rowBase = (t >> 3) << 4;
        const int colBase = (t & 7) << 4;
        const int mLane   = lane & 15;
        const int nCol    = colBase + mLane;
        const float bias  = bproj[nCol];
        const float* wrow = Wproj + (size_t)nCol * CDIM;
        v8f acc = tile_gemm(sQ, rowBase, wrow, lane, bias);
        const int rOff = (lane < 16) ? 0 : 8;
        #pragma unroll
        for (int r = 0; r < 8; ++r) {
            const int row = rowBase + rOff + r;
            out[((size_t)blk * BSZ + row) * CDIM + nCol] = acc[r];
        }
    }
}

extern "C" void kernel_launch(void* const* d_in, const int* in_sizes, int n_in,
                              void* d_out, int out_size, void* d_ws, size_t ws_size,
                              hipStream_t stream) {
    const float* x     = (const float*)d_in[0];
    const float* ef    = (const float*)d_in[1];
    const float* Wqkv  = (const float*)d_in[2];
    const float* bqkv  = (const float*)d_in[3];
    const float* Wproj = (const float*)d_in[4];
    const float* bproj = (const float*)d_in[5];
    const float* Wgate = (const float*)d_in[6];
    const float* bgate = (const float*)d_in[7];
    const int*   mask  = (const int*)d_in[8];
    (void)in_sizes; (void)n_in; (void)out_size; (void)d_ws; (void)ws_size;
    leafattn_kernel<<<NBLK, 128, 0, stream>>>(x, ef, Wqkv, bqkv, Wproj, bproj,
                                              Wgate, bgate, mask, (float*)d_out);
}